// Forward_Model_27616639714068
// MI455X (gfx1250) — compile-verified
//
#include <hip/hip_runtime.h>

// LUT-based approximate 3x3 depthwise conv, forward value only.
// out[m,c,h,w] = sum_k lut_m[ clip(round(x_nbr_k),0,255), clip(round(wprod_mk),0,255) ]
// Memory-bound: 16MB in + 32MB out => ~2.1us at 23.3 TB/s. Input is streamed
// to LDS via the CDNA5 Tensor Data Mover (tensor_load_to_lds), double-buffered
// with s_wait_tensorcnt; LUT lookups hit 18KB of LDS-resident columns.

typedef unsigned int u32;
typedef unsigned long long u64;
typedef __attribute__((ext_vector_type(4))) u32 v4u;
typedef __attribute__((ext_vector_type(8))) int v8i;
typedef __attribute__((ext_vector_type(4))) int v4i;

#define C_TOTAL   4096
#define HW        32
#define PX        (HW * HW)          // 1024 pixels / channel
#define CPT       2                  // channels per TDM tile
#define CPB       (2 * CPT)          // channels per block (2 tiles, double buffered)
#define NBLK      (C_TOTAL / CPB)    // 1024 blocks
#define NTHREADS  256                // 8 wave32
#define M_COUNT   2
#define TAPS      9

#define SM_TILE_FLOATS (2 * CPT * PX)            // two 8KB tile buffers
#define SM_LUT_OFF     SM_TILE_FLOATS
#define SM_LUT_FLOATS  (M_COUNT * TAPS * 256)    // 18KB of LUT columns
#define SM_TOTAL       (SM_TILE_FLOATS + SM_LUT_FLOATS)

#define HAS_TDM (__has_builtin(__builtin_amdgcn_tensor_load_to_lds) && \
                 __has_builtin(__builtin_amdgcn_s_wait_tensorcnt))

__device__ __forceinline__ int clamp255(float v) {
  int i = __float2int_rn(v);
  i = i < 0 ? 0 : i;
  return i > 255 ? 255 : i;
}

#if HAS_TDM
// Issue one TDM descriptor: copy nelem contiguous f32 from gsrc to LDS byte
// offset lds_byte_off. D# layout per CDNA5 ISA ch.8 (count=1, type=2 "image",
// data_size=2 (4B), 2D tile: tile_dim0=nelem, tile_dim1 unused).
__device__ __forceinline__ void tdm_issue(const float* gsrc, u32 lds_byte_off, u32 nelem) {
  const u64 ga = (u64)(const void*)gsrc;
  v4u g0 = {
    1u,                                               // [1:0] count=1, user mode
    lds_byte_off,                                     // [63:32] lds_addr (bytes)
    (u32)ga,                                          // global_addr[31:0]
    ((u32)(ga >> 32) & 0x01FFFFFFu) | (2u << 30)      // global_addr[56:32] | type=2
  };
  v8i g1 = {
    (int)(2u << 16),                                  // data_size=2 -> 4 bytes; wg_mask=0
    (int)((nelem & 0xFFFFu) << 16),                   // tensor_dim0[15:0] @ bits63:48
    (int)(((nelem >> 16) & 0xFFFFu) | (1u << 16)),    // tensor_dim0[31:16] | tensor_dim1=1
    (int)((nelem & 0xFFFFu) << 16),                   // tile_dim0 @ bits127:112
    0,                                                // tile_dim1=0, tile_dim2=0 (unused)
    (int)nelem,                                       // tensor_dim0_stride[31:0]
    0, 0
  };
  v4i z = {0, 0, 0, 0};
#if defined(__clang_major__) && (__clang_major__ >= 23)
  v8i z8 = {0, 0, 0, 0, 0, 0, 0, 0};
  __builtin_amdgcn_tensor_load_to_lds(g0, g1, z, z, z8, 0);   // clang-23 6-arg form
#else
  __builtin_amdgcn_tensor_load_to_lds(g0, g1, z, z, 0);        // ROCm 7.2 clang-22 5-arg
#endif
}
#endif

// Compute one 2-channel tile: each thread owns 4 consecutive pixels of a row
// (256 threads * 4 px = one full 32x32 channel per ch iteration).
__device__ __forceinline__ void compute_tile(const float* __restrict__ smem,
                                             int buf, int cbase, int tid,
                                             float* __restrict__ out) {
  const float* tile  = smem + buf * (CPT * PX);
  const float* lutm0 = smem + SM_LUT_OFF;
  const float* lutm1 = smem + SM_LUT_OFF + TAPS * 256;
  const int p0 = tid * 4;
  const int h  = p0 >> 5;
  const int w0 = p0 & (HW - 1);
  for (int ch = 0; ch < CPT; ++ch) {
    const float* img = tile + ch * PX;
    float a0[4] = {0.f, 0.f, 0.f, 0.f};
    float a1[4] = {0.f, 0.f, 0.f, 0.f};
#pragma unroll
    for (int dy = -1; dy <= 1; ++dy) {
      const int  hh    = h + dy;
      const bool rowok = (hh >= 0) && (hh < HW);
#pragma unroll
      for (int dx = -1; dx <= 1; ++dx) {
        const int    k  = (dy + 1) * 3 + (dx + 1);
        const float* l0 = lutm0 + k * 256;
        const float* l1 = lutm1 + k * 256;
#pragma unroll
        for (int i = 0; i < 4; ++i) {
          const int ww = w0 + i + dx;
          float v = 0.f;                               // zero 'same' padding
          if (rowok && (ww >= 0) && (ww < HW)) v = img[hh * HW + ww];
          const int ia = clamp255(v);
          a0[i] += l0[ia];                             // LDS gathers
          a1[i] += l1[ia];
        }
      }
    }
    const size_t obase = (size_t)(cbase + ch) * PX + (size_t)p0;
    *(float4*)(out + obase)                           = make_float4(a0[0], a0[1], a0[2], a0[3]);
    *(float4*)(out + (size_t)C_TOTAL * PX + obase)    = make_float4(a1[0], a1[1], a1[2], a1[3]);
  }
}

__global__ __launch_bounds__(NTHREADS)
void Forward_Model_27616639714068_kernel(const float* __restrict__ x,
                                         const float* __restrict__ weight,
                                         const float* __restrict__ wfac,
                                         const float* __restrict__ luts,
                                         float* __restrict__ out) {
  __shared__ float smem[SM_TOTAL];
  const int tid = (int)threadIdx.x;
  const int c0  = (int)blockIdx.x * CPB;

#if HAS_TDM
  // Wave 0 issues both tile DMAs up front (double buffered).
  if (tid < 32) {
    tdm_issue(x + (size_t)c0 * PX,          0u,                              CPT * PX);
    tdm_issue(x + (size_t)(c0 + CPT) * PX,  (u32)(CPT * PX * sizeof(float)), CPT * PX);
  }
#endif

  // All threads: stage the 18 needed LUT columns (m x 9 taps x 256 entries)
  // into LDS; overlaps the first tensor DMA. Column index = round(wprod).
  for (int j = tid; j < SM_LUT_FLOATS; j += NTHREADS) {
    const int m  = j / (TAPS * 256);
    const int r  = j - m * (TAPS * 256);
    const int k  = r >> 8;
    const int ia = r & 255;
    const float wc = fminf(fmaxf(weight[m * TAPS + k], 0.f), 1.f);   // hardtanh(w,0,1)
    const float wf = fminf(fmaxf(wfac[m], 1.f), 255.f);              // hardtanh(wf,1,255)
    const int   ib = clamp255(wc * wf);
    smem[SM_LUT_OFF + j] = luts[(size_t)m * 65536 + ((size_t)ia << 8) + (size_t)ib];
  }

#if !HAS_TDM
  // Fallback path (no TDM builtin): cooperative plain loads of both tiles.
  for (int j = tid; j < 2 * CPT * PX; j += NTHREADS)
    smem[j] = x[(size_t)c0 * PX + (size_t)j];
#endif

#if HAS_TDM
  if (tid < 32) __builtin_amdgcn_s_wait_tensorcnt(1);  // tile 0 landed (in-order TDM)
#endif
  __syncthreads();

  compute_tile(smem, 0, c0, tid, out);

#if HAS_TDM
  if (tid < 32) __builtin_amdgcn_s_wait_tensorcnt(0);  // tile 1 landed
#endif
  __syncthreads();

  compute_tile(smem, 1, c0 + CPT, tid, out);
}

extern "C" void kernel_launch(void* const* d_in, const int* in_sizes, int n_in,
                              void* d_out, int out_size, void* d_ws, size_t ws_size,
                              hipStream_t stream) {
  const float* x    = (const float*)d_in[0];   // [1,4096,32,32]
  const float* w    = (const float*)d_in[1];   // [2,9]
  const float* wf   = (const float*)d_in[2];   // [2,1]
  const float* luts = (const float*)d_in[3];   // [2,256,256]
  float*       out  = (float*)d_out;           // [2,1,4096,32,32]
  (void)in_sizes; (void)n_in; (void)out_size; (void)d_ws; (void)ws_size;
  hipLaunchKernelGGL(Forward_Model_27616639714068_kernel,
                     dim3(NBLK), dim3(NTHREADS), 0, stream,
                     x, w, wf, luts, out);
}